// SSIM_cal_3D_54975581389053
// MI455X (gfx1250) — compile-verified
//
#include <hip/hip_runtime.h>
#include <hip/hip_bf16.h>

typedef float v2f __attribute__((ext_vector_type(2)));
typedef float v8f __attribute__((ext_vector_type(8)));

#define WIN 7
#define NIN 160               // input spatial extent
#define O   154               // output spatial extent (160-6)
#define TTH 16                // tile height (output h per block)
#define TTW 16                // tile width  (output w per block)
#define DC  22                // output depths per chunk (7*22 = 154 exactly)
#define ZPLANES (DC + WIN - 1) // 28 input planes per chunk
#define NTHREADS 288          // 9 waves of 32

// ---------------- LDS layout (in floats) ----------------
// double-buffered X/Y halo tiles: 22 rows x 24 cols (row stride 24)
#define LB (22*24)
#define OFF_XS(b) ((b)*(2*LB))
#define OFF_YS(b) ((b)*(2*LB) + LB)
// R[k][r][n]: w-filtered fields, k=0..4, r=0..23 (rows 22..23 zero pad), n=0..15
#define OFF_R    (4*LB)
// ring[s][k][r8][lane]: 7 planes of 2D-filtered fields in WMMA D layout
#define OFF_RING (OFF_R + 5*24*16)
#define OFF_RED  (OFF_RING + 7*5*8*32)
#define LDS_FLOATS (OFF_RED + NTHREADS)   // 13280 floats = 51.9 KB

// -------- CDNA5 async global->LDS copy (ASYNCcnt path) --------
__device__ __forceinline__ void async_load_b128(unsigned lds_byte, const float* gaddr) {
    asm volatile("global_load_async_to_lds_b128 %0, %1, off"
                 :: "v"(lds_byte), "v"((unsigned long long)(size_t)gaddr)
                 : "memory");
}
__device__ __forceinline__ void wait_async_le1() {
    asm volatile("s_wait_asynccnt 0x1" ::: "memory");
}
__device__ __forceinline__ void wait_async_le0() {
    asm volatile("s_wait_asynccnt 0x0" ::: "memory");
}

// Issue one plane's halo tile (X and Y) as async B128 copies into LDS.
// 264 B128 ops total -> every one of the 9 waves issues exactly ONE
// async instruction per plane (uniform ASYNCcnt for counted waits).
__device__ __forceinline__ void issue_plane_loads(
    const float* __restrict__ X, const float* __restrict__ Y,
    int n, int z, int h0, int w0, int buf, unsigned lds_base, int t) {
    if (t < 264) {
        int field = t / 132;            // 0 = X, 1 = Y
        int rr    = (t % 132) / 6;      // halo row 0..21
        int g     = (t % 132) % 6;      // 4-float group 0..5 (24 cols)
        int h_in = h0 + rr; if (h_in > NIN - 1) h_in = NIN - 1;     // clamp (masked later)
        int wg   = w0 + 4 * g; if (wg > NIN - 4) wg = NIN - 4;      // clamp, keep 16B align
        size_t idx = ((((size_t)n * NIN) + z) * NIN + h_in) * NIN + wg;
        const float* gp = (field ? Y : X) + idx;
        unsigned ldsb = lds_base +
            4u * ((unsigned)((field ? OFF_YS(buf) : OFF_XS(buf)) + rr * 24 + 4 * g));
        async_load_b128(ldsb, gp);
    }
}

__global__ __launch_bounds__(NTHREADS)
void ssim3d_fused_kernel(const float* __restrict__ X, const float* __restrict__ Y,
                         const float* __restrict__ DR, float* __restrict__ partial) {
    __shared__ float lds[LDS_FLOATS];
    const int t    = threadIdx.x;
    const int lane = t & 31;
    const int wave = t >> 5;

    const int dchunk = blockIdx.x;            // 0..6
    const int tile   = blockIdx.y;            // 0..99
    const int n      = blockIdx.z;            // 0..3
    const int h0     = (tile / 10) * TTH;
    const int w0     = (tile % 10) * TTW;
    const int dstart = dchunk * DC;

    // low 32 bits of a generic pointer to LDS == LDS byte offset (aperture layout)
    const unsigned lds_base = (unsigned)(size_t)(const void*)&lds[0];

    // Zero the two K-pad rows of R once (WMMA propagates NaN; A is 0 there,
    // but 0*NaN would still poison the accumulator).
    for (int i = t; i < 5 * 2 * 16; i += NTHREADS) {
        int k = i / 32, rem = i % 32;
        lds[OFF_R + k * 24 * 16 + (22 + rem / 16) * 16 + (rem % 16)] = 0.0f;
    }

    const float dr = DR[n];
    const float C1 = (0.01f * dr) * (0.01f * dr);
    const float C2 = (0.03f * dr) * (0.03f * dr);
    float acc = 0.0f;

    // prefetch plane 0
    issue_plane_loads(X, Y, n, dstart, h0, w0, 0, lds_base, t);

    for (int zi = 0; zi < ZPLANES; ++zi) {
        const int buf = zi & 1;
        const bool has_next = (zi + 1 < ZPLANES);
        if (has_next)
            issue_plane_loads(X, Y, n, dstart + zi + 1, h0, w0, (zi + 1) & 1, lds_base, t);
        if (has_next) wait_async_le1(); else wait_async_le0();
        __syncthreads();   // plane zi halo tiles visible in buf

        // ---- w-filter: sliding 7-tap of the 5 moment fields -> R[5][22][16]
        {
            const int xsO = OFF_XS(buf), ysO = OFF_YS(buf);
            for (int p = t; p < 22 * 16; p += NTHREADS) {
                int r = p / 16, w = p % 16;
                float s0 = 0.f, s1 = 0.f, s2 = 0.f, s3 = 0.f, s4 = 0.f;
                int rowx = xsO + r * 24 + w;
                int rowy = ysO + r * 24 + w;
#pragma unroll
                for (int j = 0; j < WIN; ++j) {
                    float x = lds[rowx + j];
                    float y = lds[rowy + j];
                    s0 += x; s1 += y; s2 += x * x; s3 += y * y; s4 += x * y;
                }
                int rb = OFF_R + r * 16 + w;
                lds[rb + 0 * 24 * 16] = s0;
                lds[rb + 1 * 24 * 16] = s1;
                lds[rb + 2 * 24 * 16] = s2;
                lds[rb + 3 * 24 * 16] = s3;
                lds[rb + 4 * 24 * 16] = s4;
            }
        }
        __syncthreads();

        // ---- h-filter via WMMA: P(16x16) = Mh(16x24 band) x R(24x16).
        // Wave k (k<5) owns field k; full-EXEC per wave as WMMA requires.
        if (wave < 5) {
            const int k    = wave;
            const int mOrN = lane & 15;     // A: row m, B: column n
            const int hi   = lane >> 4;     // K-half selector within 4-chunk
            v8f d = {0.f, 0.f, 0.f, 0.f, 0.f, 0.f, 0.f, 0.f};
#pragma unroll
            for (int ck = 0; ck < 6; ++ck) {
                const int ka = 4 * ck + 2 * hi;
                v2f a, b;
                // band matrix Mh[m][k] = (k-m) in [0,6]
                a.x = (ka     >= mOrN && ka     <= mOrN + 6) ? 1.0f : 0.0f;
                a.y = (ka + 1 >= mOrN && ka + 1 <= mOrN + 6) ? 1.0f : 0.0f;
                // B[k][n] = R[field][k][n], per 16x16x4 f32 B striping
                b.x = lds[OFF_R + k * 24 * 16 + (ka    ) * 16 + mOrN];
                b.y = lds[OFF_R + k * 24 * 16 + (ka + 1) * 16 + mOrN];
                d = __builtin_amdgcn_wmma_f32_16x16x4_f32(
                        false, a, false, b, (short)0, d, false, false);
            }
            // spill D (VGPR r holds M=r / M=r+8, N=lane%16) into ring slot zi%7
            const int s    = zi % WIN;
            const int base = OFF_RING + ((s * 5 + k) * 8) * 32 + lane;
#pragma unroll
            for (int r8 = 0; r8 < 8; ++r8)
                lds[base + r8 * 32] = d[r8];
        }
        __syncthreads();

        // ---- depth window (sum of 7 ring slots) + SSIM + masked accumulate
        if (zi >= WIN - 1 && t < 256) {
            const int h  = t >> 4, w = t & 15;
            const int l  = w + 16 * (h >> 3);   // lane holding (M=h, N=w)
            const int r8 = h & 7;               // VGPR slot for M=h
            float sx = 0.f, sy = 0.f, sxx = 0.f, syy = 0.f, sxy = 0.f;
#pragma unroll
            for (int s = 0; s < WIN; ++s) {
                int base = OFF_RING + ((s * 5) * 8 + r8) * 32 + l;
                sx  += lds[base + 0 * 8 * 32];
                sy  += lds[base + 1 * 8 * 32];
                sxx += lds[base + 2 * 8 * 32];
                syy += lds[base + 3 * 8 * 32];
                sxy += lds[base + 4 * 8 * 32];
            }
            const float inv = 1.0f / 343.0f;        // 1/7^3
            const float cn  = 343.0f / 342.0f;      // cov_norm
            float ux = sx * inv,  uy = sy * inv;
            float uxx = sxx * inv, uyy = syy * inv, uxy = sxy * inv;
            float vx  = cn * (uxx - ux * ux);
            float vy  = cn * (uyy - uy * uy);
            float vxy = cn * (uxy - ux * uy);
            float A1 = 2.0f * ux * uy + C1;
            float A2 = 2.0f * vxy + C2;
            float B1 = ux * ux + uy * uy + C1;
            float B2 = vx + vy + C2;
            if ((h0 + h) < O && (w0 + w) < O)
                acc += (A1 * A2) / (B1 * B2);
        }
        __syncthreads();
    }

    // ---- deterministic block reduction -> per-block partial
    lds[OFF_RED + t] = (t < 256) ? acc : 0.0f;
    __syncthreads();
    for (int s = 128; s > 0; s >>= 1) {
        if (t < s) lds[OFF_RED + t] += lds[OFF_RED + t + s];
        __syncthreads();
    }
    if (t == 0) {
        int bid = (n * 100 + tile) * 7 + dchunk;   // = n*700 + tile*7 + dchunk
        partial[bid] = lds[OFF_RED];
    }
}

__global__ __launch_bounds__(256)
void ssim3d_finalize(const float* __restrict__ partial, float* __restrict__ out) {
    __shared__ float red[256];
    const int b = blockIdx.x, t = threadIdx.x;
    float a = 0.0f;
    for (int i = t; i < 700; i += 256) a += partial[b * 700 + i];
    red[t] = a;
    __syncthreads();
    for (int s = 128; s > 0; s >>= 1) {
        if (t < s) red[t] += red[t + s];
        __syncthreads();
    }
    if (t == 0) out[b] = red[0] * (1.0f / (154.0f * 154.0f * 154.0f));
}

extern "C" void kernel_launch(void* const* d_in, const int* in_sizes, int n_in,
                              void* d_out, int out_size, void* d_ws, size_t ws_size,
                              hipStream_t stream) {
    (void)in_sizes; (void)n_in; (void)out_size; (void)ws_size;
    const float* X  = (const float*)d_in[0];
    const float* Y  = (const float*)d_in[1];
    const float* DR = (const float*)d_in[2];
    float* out      = (float*)d_out;
    float* partial  = (float*)d_ws;     // 2800 floats of scratch

    dim3 grid(7, 100, 4);               // d-chunks x hw-tiles x batch
    ssim3d_fused_kernel<<<grid, NTHREADS, 0, stream>>>(X, Y, DR, partial);
    ssim3d_finalize<<<4, 256, 0, stream>>>(partial, out);
}